// GATModel_72249939853876
// MI455X (gfx1250) — compile-verified
//
#include <hip/hip_runtime.h>
#include <math.h>

typedef __attribute__((ext_vector_type(2))) float v2f;
typedef __attribute__((ext_vector_type(8))) float v8f;

#define N_NODES 50000
#define C_IN    128
#define C_HID   256      // 4 heads * 64
#define HEADS   4
#define EMBED   64
#define E0      800000
#define ETOT    (E0 + N_NODES)   // with self loops
#define NEG_SLOPE 0.2f

// ---------------------------------------------------------------------------
// f32 WMMA GEMM: H[M,N] = X[M,K] @ W[K,N]
// One wave -> 16 rows x 64 cols (4 accumulators). 4 waves/block -> 16x256.
// M, K, N all multiples of 16/4 here, so no predication (EXEC all ones).
// ---------------------------------------------------------------------------
__global__ void gemm_wmma(const float* __restrict__ X,
                          const float* __restrict__ W,
                          float* __restrict__ H,
                          int K, int N) {
    const int lane   = threadIdx.x & 31;
    const int wave   = threadIdx.x >> 5;          // 0..3
    const int tileM  = blockIdx.x;                // 16 rows
    const int lane15 = lane & 15;
    const int khalf  = lane >> 4;                 // 0 or 1
    const int row    = tileM * 16 + lane15;       // A-matrix row for this lane
    const int col0   = wave * 64;

    v8f acc0 = {}, acc1 = {}, acc2 = {}, acc3 = {};

    for (int k = 0; k < K; k += 4) {
        const int kk = k + khalf * 2;             // lanes 0-15: K=k,k+1 ; 16-31: K=k+2,k+3
        v2f a;
        a.x = X[row * K + kk];
        a.y = X[row * K + kk + 1];

        const float* wp0 = W + (size_t)kk * N;
        const float* wp1 = wp0 + N;

        v2f b0, b1, b2, b3;
        b0.x = wp0[col0 +  0 + lane15]; b0.y = wp1[col0 +  0 + lane15];
        b1.x = wp0[col0 + 16 + lane15]; b1.y = wp1[col0 + 16 + lane15];
        b2.x = wp0[col0 + 32 + lane15]; b2.y = wp1[col0 + 32 + lane15];
        b3.x = wp0[col0 + 48 + lane15]; b3.y = wp1[col0 + 48 + lane15];

        acc0 = __builtin_amdgcn_wmma_f32_16x16x4_f32(false, a, false, b0, (short)0, acc0, false, false);
        acc1 = __builtin_amdgcn_wmma_f32_16x16x4_f32(false, a, false, b1, (short)0, acc1, false, false);
        acc2 = __builtin_amdgcn_wmma_f32_16x16x4_f32(false, a, false, b2, (short)0, acc2, false, false);
        acc3 = __builtin_amdgcn_wmma_f32_16x16x4_f32(false, a, false, b3, (short)0, acc3, false, false);
    }

    // C/D layout: lanes 0-15 hold M = 0..7 (VGPR i -> M=i), lanes 16-31 hold M = 8..15
    const int rbase = tileM * 16 + khalf * 8;
#pragma unroll
    for (int i = 0; i < 8; ++i) {
        float* op = H + (size_t)(rbase + i) * N + col0 + lane15;
        op[ 0] = acc0[i];
        op[16] = acc1[i];
        op[32] = acc2[i];
        op[48] = acc3[i];
    }
}

// ---------------------------------------------------------------------------
// Per (node, head) attention logits: a_src[n,h] = <h[n,h,:], att_src[h,:]>
// One wave32 per (node, head); shuffle reduction.
// ---------------------------------------------------------------------------
__global__ void attn_dots(const float* __restrict__ H,
                          const float* __restrict__ att_src,
                          const float* __restrict__ att_dst,
                          float* __restrict__ a_src,
                          float* __restrict__ a_dst) {
    const int gw   = (int)((blockIdx.x * blockDim.x + threadIdx.x) >> 5);
    const int lane = threadIdx.x & 31;
    const int node = gw >> 2;
    const int head = gw & 3;
    if (node >= N_NODES) return;

    const float* hp = H + (size_t)node * C_HID + head * EMBED;
    const float* as = att_src + head * EMBED;
    const float* ad = att_dst + head * EMBED;

    float s = 0.f, d = 0.f;
#pragma unroll
    for (int c = 0; c < EMBED; c += 32) {
        float v = hp[c + lane];
        s += v * as[c + lane];
        d += v * ad[c + lane];
    }
#pragma unroll
    for (int off = 16; off > 0; off >>= 1) {
        s += __shfl_down(s, off, 32);
        d += __shfl_down(d, off, 32);
    }
    if (lane == 0) {
        a_src[node * HEADS + head] = s;
        a_dst[node * HEADS + head] = d;
    }
}

// ---------------------------------------------------------------------------
// order-preserving float <-> uint encoding for atomic max
// ---------------------------------------------------------------------------
__device__ __forceinline__ unsigned ford(float f) {
    unsigned u = __float_as_uint(f);
    return (u >> 31) ? ~u : (u | 0x80000000u);
}
__device__ __forceinline__ float funord(unsigned u) {
    return (u >> 31) ? __uint_as_float(u ^ 0x80000000u) : __uint_as_float(~u);
}

__global__ void fill_u32(unsigned* __restrict__ p, unsigned v, int n) {
    int i = blockIdx.x * blockDim.x + threadIdx.x;
    if (i < n) p[i] = v;
}

__device__ __forceinline__ void edge_sd(int e, const int* src, const int* dst, int& s, int& d) {
    if (e < E0) { s = src[e]; d = dst[e]; }
    else        { s = e - E0; d = s; }
}

__device__ __forceinline__ float edge_logit(const float* a_src, const float* a_dst,
                                            int s, int d, int h) {
    float v = a_src[s * HEADS + h] + a_dst[d * HEADS + h];
    return v > 0.f ? v : NEG_SLOPE * v;   // LeakyReLU(0.2)
}

// pass 1: per-dst per-head running max (segment_max)
__global__ void edge_max(const int* __restrict__ src, const int* __restrict__ dst,
                         const float* __restrict__ a_src, const float* __restrict__ a_dst,
                         unsigned* __restrict__ m) {
    int i = blockIdx.x * blockDim.x + threadIdx.x;
    if (i >= ETOT * HEADS) return;
    int e = i >> 2, h = i & 3, s, d;
    edge_sd(e, src, dst, s, d);
    float v = edge_logit(a_src, a_dst, s, d, h);
    atomicMax(&m[d * HEADS + h], ford(v));
}

// pass 2: ee = exp(e - m[dst]); z[dst] += ee  (segment_sum of exps)
__global__ void edge_expsum(const int* __restrict__ src, const int* __restrict__ dst,
                            const float* __restrict__ a_src, const float* __restrict__ a_dst,
                            const unsigned* __restrict__ m,
                            float* __restrict__ ee, float* __restrict__ z) {
    int i = blockIdx.x * blockDim.x + threadIdx.x;
    if (i >= ETOT * HEADS) return;
    int e = i >> 2, h = i & 3, s, d;
    edge_sd(e, src, dst, s, d);
    float v  = edge_logit(a_src, a_dst, s, d, h);
    float mv = funord(m[d * HEADS + h]);
    float ev = __expf(v - mv);
    ee[i] = ev;
    atomicAdd(&z[d * HEADS + h], ev);
}

// pass 3: agg[dst] += alpha * h[src]  -- one 256-thread block per edge
__global__ void edge_scatter(const int* __restrict__ src, const int* __restrict__ dst,
                             const float* __restrict__ ee, const float* __restrict__ z,
                             const float* __restrict__ H, float* __restrict__ agg) {
    int e = blockIdx.x;
    int t = threadIdx.x;            // 0..255 -> head = t>>6, c = t&63
    int h = t >> 6, s, d;
    edge_sd(e, src, dst, s, d);
    float alpha = ee[(size_t)e * HEADS + h] / (z[d * HEADS + h] + 1e-16f);
    float val   = H[(size_t)s * C_HID + t] * alpha;
    atomicAdd(&agg[(size_t)d * C_HID + t], val);
}

// bias (+ optional ReLU)
__global__ void bias_act(const float* __restrict__ agg, const float* __restrict__ bias,
                         float* __restrict__ out, int n, int relu) {
    int i = blockIdx.x * blockDim.x + threadIdx.x;
    if (i >= n) return;
    float v = agg[i] + bias[i & (C_HID - 1)];
    out[i] = relu ? fmaxf(v, 0.f) : v;
}

// ---------------------------------------------------------------------------
extern "C" void kernel_launch(void* const* d_in, const int* in_sizes, int n_in,
                              void* d_out, int out_size, void* d_ws, size_t ws_size,
                              hipStream_t stream) {
    const float* x        = (const float*)d_in[0];
    const int*   eidx     = (const int*)  d_in[1];   // [2, E0] flattened
    const float* W1       = (const float*)d_in[2];
    const float* att_s1   = (const float*)d_in[3];
    const float* att_d1   = (const float*)d_in[4];
    const float* b1       = (const float*)d_in[5];
    const float* W2       = (const float*)d_in[6];
    const float* att_s2   = (const float*)d_in[7];
    const float* att_d2   = (const float*)d_in[8];
    const float* b2       = (const float*)d_in[9];
    float*       out      = (float*)d_out;

    const int* src = eidx;
    const int* dst = eidx + E0;

    // workspace carve-up (floats)
    const size_t NB = (size_t)N_NODES * C_HID;       // 12.8M
    float* h      = (float*)d_ws;                    // GEMM output (both layers)
    float* agg    = h      + NB;                     // aggregation buffer
    float* hrelu  = agg    + NB;                     // layer-1 activated output
    float* ee     = hrelu  + NB;                     // per-edge exp [ETOT*4]
    float* a_src  = ee     + (size_t)ETOT * HEADS;
    float* a_dst  = a_src  + (size_t)N_NODES * HEADS;
    unsigned* m   = (unsigned*)(a_dst + (size_t)N_NODES * HEADS);
    float* z      = (float*)(m + (size_t)N_NODES * HEADS);

    const int NH       = N_NODES * HEADS;            // 200000
    const int EH       = ETOT * HEADS;               // 3.4M
    const int NELEM    = (int)NB;                    // 12.8M
    const unsigned ORD_NEG_INF = 0x007FFFFFu;        // ford(-inf)

    const dim3 B256(256);
    const int gNH  = (NH    + 255) / 256;
    const int gEH  = (EH    + 255) / 256;
    const int gNE  = (NELEM + 255) / 256;
    const int gMM  = N_NODES / 16;                   // 3125
    const int gDOT = (NH * 32 + 255) / 256;          // wave per (node,head)

    // ---------------- layer 1 ----------------
    gemm_wmma  <<<gMM, 128, 0, stream>>>(x, W1, h, C_IN, C_HID);
    attn_dots  <<<gDOT, B256, 0, stream>>>(h, att_s1, att_d1, a_src, a_dst);
    fill_u32   <<<gNH, B256, 0, stream>>>(m, ORD_NEG_INF, NH);
    fill_u32   <<<gNH, B256, 0, stream>>>((unsigned*)z, 0u, NH);
    fill_u32   <<<gNE, B256, 0, stream>>>((unsigned*)agg, 0u, NELEM);
    edge_max   <<<gEH, B256, 0, stream>>>(src, dst, a_src, a_dst, m);
    edge_expsum<<<gEH, B256, 0, stream>>>(src, dst, a_src, a_dst, m, ee, z);
    edge_scatter<<<ETOT, B256, 0, stream>>>(src, dst, ee, z, h, agg);
    bias_act   <<<gNE, B256, 0, stream>>>(agg, b1, hrelu, NELEM, 1);

    // ---------------- layer 2 ----------------
    gemm_wmma  <<<gMM, 128, 0, stream>>>(hrelu, W2, h, C_HID, C_HID);
    attn_dots  <<<gDOT, B256, 0, stream>>>(h, att_s2, att_d2, a_src, a_dst);
    fill_u32   <<<gNH, B256, 0, stream>>>(m, ORD_NEG_INF, NH);
    fill_u32   <<<gNH, B256, 0, stream>>>((unsigned*)z, 0u, NH);
    fill_u32   <<<gNE, B256, 0, stream>>>((unsigned*)agg, 0u, NELEM);
    edge_max   <<<gEH, B256, 0, stream>>>(src, dst, a_src, a_dst, m);
    edge_expsum<<<gEH, B256, 0, stream>>>(src, dst, a_src, a_dst, m, ee, z);
    edge_scatter<<<ETOT, B256, 0, stream>>>(src, dst, ee, z, h, agg);
    bias_act   <<<gNE, B256, 0, stream>>>(agg, b2, out, NELEM, 0);
}